// SCM_29721173688893
// MI455X (gfx1250) — compile-verified
//
#include <hip/hip_runtime.h>

// out = (eps @ inv(I - A)) * w + b
//   1) Nm = I - A, X0 = I + A
//   2) 5x Newton-Schulz: T = Nm@X ; X' = 2X - X@T   (||A||~0.64 -> residual A^64 ~ 4e-13)
//   3) out = (eps @ X) * w + b                       (fused affine epilogue)
// GEMM: V_WMMA_F32_16X16X4_F32, 128x128 macro-tile, 8 waves (2x4), 64x32 per wave.
// Staging: Tensor Data Mover (tensor_load_to_lds, 6-arg builtin) double-buffered,
// TENSORcnt-synced; TDM pad_enable recreates LDS padding (A stride 18, B stride 130).

typedef __attribute__((ext_vector_type(2))) float v2f;
typedef __attribute__((ext_vector_type(4))) float v4f;
typedef __attribute__((ext_vector_type(8))) float v8f;
typedef __attribute__((ext_vector_type(4))) unsigned int u32x4;
typedef __attribute__((ext_vector_type(8))) int i32x8;
typedef __attribute__((ext_vector_type(4))) int i32x4;

#define BM 128
#define BN 128
#define BK 16
#define LDA (BK + 2)   // A tile LDS row stride: 16 data + 2 pad (TDM pad_interval=16dw, amount=2dw)
#define LDB (BN + 2)   // B tile LDS row stride: 128 data + 2 pad (pad_interval=128dw, amount=2dw)

#define HAVE_TDM __has_builtin(__builtin_amdgcn_tensor_load_to_lds)

#if HAVE_TDM
// Issue a 2D-tile TDM load (CDNA5 ISA). data_size = 4 bytes.
// pad_icode: 3 -> pad every 16 DWORDs, 6 -> every 128 DWORDs; pad amount code 1 -> 2 DWORDs.
__device__ __forceinline__ void tdm_load_2d(unsigned lds_off, const float* gptr,
                                            unsigned tensor_d0, unsigned tensor_d1,
                                            unsigned tile_d0, unsigned tile_d1,
                                            unsigned stride0_elems, unsigned pad_icode)
{
    unsigned long long ga = (unsigned long long)(size_t)gptr;
    u32x4 g0;
    g0[0] = 1u;                                          // count=1, no gather/iterate
    g0[1] = lds_off;                                     // lds_addr (bytes)
    g0[2] = (unsigned)(ga & 0xFFFFFFFFu);                // global_addr[31:0]
    g0[3] = (unsigned)((ga >> 32) & 0x01FFFFFFu)         // global_addr[56:32]
          | (2u << 30);                                  // type = 2 ("image")
    i32x8 g1;
    g1[0] = (int)((2u << 16)                             // data_size = 4B
                | (1u << 20)                             // pad_enable
                | (pad_icode << 22)                      // pad_interval
                | (1u << 25));                           // pad_amount code 1 -> 2 DWORDs
    g1[1] = (int)((tensor_d0 & 0xFFFFu) << 16);          // abar_addr=0 | tensor_dim0[15:0]
    g1[2] = (int)(((tensor_d0 >> 16) & 0xFFFFu) | ((tensor_d1 & 0xFFFFu) << 16));
    g1[3] = (int)(((tensor_d1 >> 16) & 0xFFFFu) | ((tile_d0 & 0xFFFFu) << 16));
    g1[4] = (int)(tile_d1 & 0xFFFFu);                    // tile_dim1 | tile_dim2 = 0
    g1[5] = (int)stride0_elems;                          // tensor_dim0_stride[31:0]
    g1[6] = 0;                                           // stride0[47:32] | stride1[15:0]
    g1[7] = 0;
    i32x4 z4 = {0, 0, 0, 0};                             // 2D tile: groups 2/3 unused
    i32x8 z8 = {0, 0, 0, 0, 0, 0, 0, 0};                 // trailing group (clang-23 6-arg form)
    __builtin_amdgcn_tensor_load_to_lds(g0, g1, z4, z4, z8, 0);
}
#endif

// mode 0: Out = alpha*(A@B) + beta*Cin       mode 1: Out = (A@B)*wv[col] + bv[col]
__global__ __launch_bounds__(256)
void gemm_wmma_f32(const float* __restrict__ A, const float* __restrict__ B,
                   const float* __restrict__ Cin, float* __restrict__ Out,
                   int M, int N, int K, float alpha, float beta,
                   const float* __restrict__ wv, const float* __restrict__ bv,
                   int mode)
{
    __shared__ float As[2][BM * LDA];   // row-major [m][k], stride 18
    __shared__ float Bs[2][BK * LDB];   // row-major [k][n], stride 130

    const int tid  = threadIdx.x;
    const int lane = tid & 31;
    const int wave = tid >> 5;
    const int waveRow = wave >> 2;      // 0..1 : 64-row half
    const int waveCol = wave & 3;       // 0..3 : 32-col quarter

    const int blockRow = blockIdx.y * BM;
    const int blockCol = blockIdx.x * BN;

    const int lhalf = lane >> 4;        // K-pair select (A,B frags) / row+8 (C frag)
    const int lmod  = lane & 15;        // M (A-frag), N (B/C-frag)

    v8f acc[4][2] = {};

    const int nstages = K / BK;

#if HAVE_TDM
    if (wave == 0) {
        tdm_load_2d((unsigned)(size_t)&As[0][0], A + (size_t)blockRow * K,
                    K, M - blockRow, BK, BM, K, 3);
        tdm_load_2d((unsigned)(size_t)&Bs[0][0], B + blockCol,
                    N - blockCol, K, BN, BK, N, 6);
    }
#endif

    for (int s = 0; s < nstages; ++s) {
        const int buf = s & 1;
        const int k0 = s * BK;

#if HAVE_TDM
        __builtin_amdgcn_s_wait_tensorcnt(0);   // stage-s tiles landed (wave0; others cnt==0)
        __syncthreads();
        if (wave == 0 && s + 1 < nstages) {     // prefetch stage s+1 while computing stage s
            const int k1 = k0 + BK;
            tdm_load_2d((unsigned)(size_t)&As[buf ^ 1][0], A + (size_t)blockRow * K + k1,
                        K - k1, M - blockRow, BK, BM, K, 3);
            tdm_load_2d((unsigned)(size_t)&Bs[buf ^ 1][0], B + (size_t)k1 * N + blockCol,
                        N - blockCol, K - k1, BN, BK, N, 6);
        }
#else
        // Fallback: manual vector-load staging (128x16 and 16x128 tiles, 2 float4/thread each)
        #pragma unroll
        for (int t = 0; t < 2; ++t) {
            int idx = tid + t * 256;                 // 0..511
            int row = idx >> 2;                      // 4 float4 per row
            int kq  = (idx & 3) << 2;
            v4f v = *(const v4f*)(A + (size_t)(blockRow + row) * K + k0 + kq);
            float* p = &As[buf][row * LDA + kq];
            p[0] = v[0]; p[1] = v[1]; p[2] = v[2]; p[3] = v[3];
        }
        #pragma unroll
        for (int t = 0; t < 2; ++t) {
            int idx = tid + t * 256;                 // 0..511
            int kk  = idx >> 5;                      // 32 float4 per row
            int nq  = (idx & 31) << 2;
            v4f v = *(const v4f*)(B + (size_t)(k0 + kk) * N + blockCol + nq);
            float* p = &Bs[buf][kk * LDB + nq];
            p[0] = v[0]; p[1] = v[1]; p[2] = v[2]; p[3] = v[3];
        }
        __syncthreads();
#endif

        const float* Asb = &As[buf][0];
        const float* Bsb = &Bs[buf][0];
        #pragma unroll
        for (int kb = 0; kb < BK; kb += 4) {
            // A frag 16x4: lane<16 -> (M=lmod, K=kb+{0,1}); lane>=16 -> K=kb+{2,3}
            v2f afrag[4];
            #pragma unroll
            for (int fr = 0; fr < 4; ++fr)
                afrag[fr] = *(const v2f*)&Asb[(waveRow * 64 + fr * 16 + lmod) * LDA
                                              + kb + lhalf * 2];
            // B frag 4x16: lane<16 -> (K=kb+{0,1}, N=lmod); lane>=16 -> K=kb+{2,3}
            v2f bfrag[2];
            #pragma unroll
            for (int fc = 0; fc < 2; ++fc) {
                const float* p = &Bsb[(kb + lhalf * 2) * LDB + waveCol * 32 + fc * 16 + lmod];
                bfrag[fc][0] = p[0];
                bfrag[fc][1] = p[LDB];
            }
            #pragma unroll
            for (int fr = 0; fr < 4; ++fr)
                #pragma unroll
                for (int fc = 0; fc < 2; ++fc)
                    acc[fr][fc] = __builtin_amdgcn_wmma_f32_16x16x4_f32(
                        false, afrag[fr], false, bfrag[fc],
                        (short)0, acc[fr][fc], false, false);
        }
        __syncthreads();   // compute done: buffer may be re-DMA'd next iteration
    }

    // C frag: VGPR r -> row (r + 8*lhalf), col lmod
    const int m0 = blockRow + waveRow * 64;
    const int n0 = blockCol + waveCol * 32;
    #pragma unroll
    for (int fr = 0; fr < 4; ++fr) {
        #pragma unroll
        for (int fc = 0; fc < 2; ++fc) {
            const int col = n0 + fc * 16 + lmod;
            const int rowBase = m0 + fr * 16 + lhalf * 8;
            if (mode == 1) {
                const float wc = wv[col];
                const float bc = bv[col];
                #pragma unroll
                for (int r = 0; r < 8; ++r)
                    Out[(size_t)(rowBase + r) * N + col] = acc[fr][fc][r] * wc + bc;
            } else {
                #pragma unroll
                for (int r = 0; r < 8; ++r) {
                    const size_t o = (size_t)(rowBase + r) * N + col;
                    float v = alpha * acc[fr][fc][r];
                    if (beta != 0.0f) v += beta * Cin[o];
                    Out[o] = v;
                }
            }
        }
    }
}

// Nm = I - A ; X = I + A  (Newton-Schulz seed: residual = A^2)
__global__ __launch_bounds__(256)
void init_nm_x(const float* __restrict__ A, float* __restrict__ Nm,
               float* __restrict__ X, int D)
{
    size_t i = (size_t)blockIdx.x * 256 + threadIdx.x;
    if (i >= (size_t)D * D) return;
    int r = (int)(i / D);
    int c = (int)(i % D);
    float a  = A[i];
    float id = (r == c) ? 1.0f : 0.0f;
    Nm[i] = id - a;
    X[i]  = id + a;
}

extern "C" void kernel_launch(void* const* d_in, const int* in_sizes, int n_in,
                              void* d_out, int out_size, void* d_ws, size_t ws_size,
                              hipStream_t stream)
{
    const float* eps = (const float*)d_in[0];   // [Brows, D]
    const float* A   = (const float*)d_in[1];   // [D, D]
    const float* w   = (const float*)d_in[2];   // [D]
    const float* b   = (const float*)d_in[3];   // [D]

    const int D     = in_sizes[2];
    const int Brows = in_sizes[0] / D;

    float* Nm = (float*)d_ws;                   // 4 * D*D floats = 16 MB workspace
    float* X0 = Nm + (size_t)D * D;
    float* X1 = X0 + (size_t)D * D;
    float* T  = X1 + (size_t)D * D;

    init_nm_x<<<(D * D + 255) / 256, 256, 0, stream>>>(A, Nm, X0, D);

    dim3 blk(256);
    dim3 grdD(D / BN, D / BM);
    float* cur = X0;
    float* nxt = X1;
    for (int it = 0; it < 5; ++it) {
        gemm_wmma_f32<<<grdD, blk, 0, stream>>>(Nm, cur, nullptr, T,
                                                D, D, D, 1.0f, 0.0f,
                                                nullptr, nullptr, 0);
        gemm_wmma_f32<<<grdD, blk, 0, stream>>>(cur, T, cur, nxt,
                                                D, D, D, -1.0f, 2.0f,
                                                nullptr, nullptr, 0);
        float* tmp = cur; cur = nxt; nxt = tmp;
    }

    dim3 grdO(D / BN, Brows / BM);
    gemm_wmma_f32<<<grdO, blk, 0, stream>>>(eps, cur, nullptr, (float*)d_out,
                                            Brows, D, D, 1.0f, 0.0f,
                                            w, b, 1);
}